// TransmissionEstimator_43645457662631
// MI455X (gfx1250) — compile-verified
//
#include <hip/hip_runtime.h>
#include <hip/hip_bf16.h>
#include <stdint.h>

#define N_IMG 8
#define CH    3
#define H     1024
#define W     1024
#define HW    (H*W)
#define PAD   7
#define WIN   15
#define KSEL  1048              /* int(1024*1024*0.001) */
#define NBINS 4096
#define A_MAX_F (220.0f/255.0f)

/* pass-2 tiling: 32 output rows x 256 cols, 46 staged rows */
#define P2_TH   32
#define P2_TW   256
#define P2_ROWS (P2_TH + WIN - 1)
#define HALO_W  (P2_TW + WIN - 1)   /* 270 */

typedef unsigned int v4u  __attribute__((ext_vector_type(4)));
typedef int          v8i_ __attribute__((ext_vector_type(8)));
typedef int          v4i_ __attribute__((ext_vector_type(4)));

__device__ __forceinline__ float fmin3(float a, float b, float c) {
  return fminf(a, fminf(b, c));
}

#if defined(__gfx1250__) && __has_builtin(__builtin_amdgcn_tensor_load_to_lds)
#define HAVE_TDM 1
/* 2-D f32 tile DMA: rows x cols, global row stride = stride_elems, packed in LDS */
__device__ __forceinline__ void tdm_load_2d_f32(unsigned lds_addr, const void* gptr,
                                                unsigned cols, unsigned rows,
                                                unsigned stride_elems) {
  unsigned long long ga = (unsigned long long)(uintptr_t)gptr;
  v4u g0;
  g0.x = 1u;                                   /* count=1, user load          */
  g0.y = lds_addr;                             /* lds_addr (bytes)            */
  g0.z = (unsigned)(ga & 0xFFFFFFFFu);         /* global_addr[31:0]           */
  g0.w = (unsigned)((ga >> 32) & 0x1FFFFFFu)   /* global_addr[56:32]          */
       | (2u << 30);                           /* type = 2 (image)            */
  v8i_ g1;
  g1[0] = (int)(2u << 16);                     /* data_size=2 -> 4 bytes      */
  g1[1] = (int)(cols << 16);                   /* tensor_dim0[15:0]           */
  g1[2] = (int)(rows << 16);                   /* dim0 hi=0 | tensor_dim1 lo  */
  g1[3] = (int)(cols << 16);                   /* dim1 hi=0 | tile_dim0       */
  g1[4] = (int)rows;                           /* tile_dim1 | tile_dim2=0     */
  g1[5] = (int)stride_elems;                   /* tensor_dim0_stride lo32     */
  g1[6] = 0;
  g1[7] = 0;
  v4i_ gz4 = {0, 0, 0, 0};
  v8i_ gz8 = {0, 0, 0, 0, 0, 0, 0, 0};
  __builtin_amdgcn_tensor_load_to_lds(g0, g1, gz4, gz4, gz8, 0);
}
#endif

/* ---- pass 2: vertical 15-min (LDS tile via TDM) + dc write + histogram ---- */
__global__ void k_colmin_hist(const float* __restrict__ rowmin,
                              float* __restrict__ dc,
                              unsigned* __restrict__ ghist) {
  __shared__ float    tile[P2_ROWS * P2_TW];  /* 46*256*4 = 47104 B */
  __shared__ unsigned lhist[NBINS];           /* 16384 B */
  int b   = blockIdx.x;
  int n   = b >> 7;             /* 128 blocks per image: 32 ytiles * 4 xtiles */
  int rem = b & 127;
  int y0  = (rem >> 2) * P2_TH;
  int x0  = (rem & 3) * P2_TW;
  int t   = threadIdx.x;

  int ytop = y0 - PAD;
  int y_lo = ytop < 0 ? 0 : ytop;
  int y_hi = ytop + P2_ROWS; if (y_hi > H) y_hi = H;
  int rows = y_hi - y_lo;
  const float* gsrc = rowmin + ((size_t)n * H + y_lo) * W + x0;

#ifdef HAVE_TDM
  if (t < 32) {   /* wave 0 issues one TDM DMA for the whole tile (overlaps below) */
    unsigned lds_base = (unsigned)(unsigned long long)(uintptr_t)
                        (&tile[(y_lo - ytop) * P2_TW]);
    tdm_load_2d_f32(lds_base, gsrc, P2_TW, (unsigned)rows, W);
  }
#endif

  for (int i = t; i < NBINS; i += 256) lhist[i] = 0u;
  /* pad rows outside the image with 1.0 */
  for (int r = 0; r < P2_ROWS; ++r) {
    int gy = ytop + r;
    if (gy < 0 || gy >= H) tile[r * P2_TW + t] = 1.0f;
  }

#ifdef HAVE_TDM
  if (t < 32) __builtin_amdgcn_s_wait_tensorcnt(0);
#else
  for (int r = 0; r < rows; ++r)
    tile[(y_lo - ytop + r) * P2_TW + t] = gsrc[(size_t)r * W + t];
#endif
  __syncthreads();

  float* dcb = dc + ((size_t)n * H + y0) * W + x0 + t;
#pragma unroll 1
  for (int oy = 0; oy < P2_TH; ++oy) {
    float m = tile[oy * P2_TW + t];
#pragma unroll
    for (int j = 1; j < WIN; ++j) m = fminf(m, tile[(oy + j) * P2_TW + t]);
    dcb[(size_t)oy * W] = m;
    int bin = (int)(m * (float)NBINS);
    if (bin > NBINS - 1) bin = NBINS - 1;
    if (bin < 0) bin = 0;
    atomicAdd(&lhist[bin], 1u);
  }
  __syncthreads();
  unsigned* gh = ghist + n * NBINS;
  for (int i = t; i < NBINS; i += 256) {
    unsigned v = lhist[i];
    if (v) atomicAdd(&gh[i], v);
  }
}

/* ---- pass 1: channel-min + horizontal 15-wide min (pad 1.0), TDM staged ---- */
__global__ void k_rowmin(const float* __restrict__ I, float* __restrict__ rowmin) {
  __shared__ float chbuf[CH * HALO_W];      /* 3 x <=270 packed rows */
  __shared__ float cmin[HALO_W];
  int b   = blockIdx.x;
  int n   = b >> 12;            /* 4096 blocks per image: 1024 rows * 4 tiles */
  int rem = b & 4095;
  int y   = rem >> 2;
  int x0  = (rem & 3) << 8;
  int t   = threadIdx.x;

  int xl   = x0 - PAD;
  int x_lo = xl < 0 ? 0 : xl;
  int x_hi = x0 + P2_TW + PAD - 1; if (x_hi > W - 1) x_hi = W - 1;
  int cols = x_hi - x_lo + 1;   /* 263..270 */
  int off  = x_lo - xl;         /* 0 or 7   */
  const float* gsrc = I + (size_t)n * CH * HW + (size_t)y * W + x_lo;

#ifdef HAVE_TDM
  if (t < 32) {   /* one 2-D DMA: 3 channel rows, global row stride = HW */
    unsigned lds_base = (unsigned)(unsigned long long)(uintptr_t)(&chbuf[0]);
    tdm_load_2d_f32(lds_base, gsrc, (unsigned)cols, CH, HW);
    __builtin_amdgcn_s_wait_tensorcnt(0);
  }
#else
  for (int c = 0; c < CH; ++c)
    for (int i = t; i < cols; i += 256)
      chbuf[c * cols + i] = gsrc[(size_t)c * HW + i];
#endif
  __syncthreads();

  for (int i = t; i < HALO_W; i += 256) {
    int gx = xl + i;
    float v = 1.0f;
    if (gx >= x_lo && gx <= x_hi) {
      int j = i - off;
      v = fmin3(chbuf[j], chbuf[cols + j], chbuf[2 * cols + j]);
    }
    cmin[i] = v;
  }
  __syncthreads();
  float m = cmin[t];
#pragma unroll
  for (int j = 1; j < WIN; ++j) m = fminf(m, cmin[t + j]);
  rowmin[((size_t)n * H + y) * W + x0 + t] = m;
}

/* ---- init: zero per-image histograms and channel max accumulators ---- */
__global__ void k_init(unsigned* __restrict__ hist, unsigned* __restrict__ amax) {
  int i = blockIdx.x * blockDim.x + threadIdx.x;
  if (i < N_IMG * NBINS) hist[i] = 0u;
  if (i < N_IMG * CH)    amax[i] = 0u;
}

/* ---- pass 3: per-image threshold = lower edge of bin holding k-th largest ---- */
__global__ void k_thresh(const unsigned* __restrict__ ghist, float* __restrict__ thresh) {
  __shared__ unsigned csum[256];
  int n = blockIdx.x;
  int t = threadIdx.x;
  const unsigned* h = ghist + n * NBINS;
  unsigned s = 0;
  for (int j = 0; j < 16; ++j) s += h[t * 16 + j];
  csum[t] = s;
  __syncthreads();
  if (t == 0) {
    unsigned run = 0;
    int c;
    int found = 0;
    for (c = 255; c >= 0; --c) {
      if (run + csum[c] >= (unsigned)KSEL) break;
      run += csum[c];
    }
    if (c >= 0) {
      int bb;
      for (bb = c * 16 + 15; bb >= c * 16; --bb) {
        run += h[bb];
        if (run >= (unsigned)KSEL) break;
      }
      found = (bb < c * 16) ? c * 16 : bb;
    }
    thresh[n] = (float)found / (float)NBINS;
  }
}

/* ---- pass 4: atomic-max I over pixels with dc >= threshold ---- */
__global__ void k_select(const float* __restrict__ I, const float* __restrict__ dc,
                         const float* __restrict__ thresh, unsigned* __restrict__ amax) {
  size_t idx = (size_t)blockIdx.x * 256 + threadIdx.x;
  int    n   = (int)(idx >> 20);
  size_t pix = idx & (size_t)(HW - 1);
  float v = dc[idx];
  if (v >= thresh[n]) {
    const float* ib = I + (size_t)n * CH * HW + pix;
#pragma unroll
    for (int c = 0; c < CH; ++c)
      atomicMax(&amax[n * CH + c], __float_as_uint(ib[(size_t)c * HW]));
  }
}

/* ---- pass 5: clamp + emit [8,3] ---- */
__global__ void k_final(const unsigned* __restrict__ amax, float* __restrict__ out) {
  int i = threadIdx.x;
  if (i < N_IMG * CH) out[i] = fminf(__uint_as_float(amax[i]), A_MAX_F);
}

extern "C" void kernel_launch(void* const* d_in, const int* in_sizes, int n_in,
                              void* d_out, int out_size, void* d_ws, size_t ws_size,
                              hipStream_t stream) {
  const float* I = (const float*)d_in[0];
  float* out = (float*)d_out;
  char* ws = (char*)d_ws;

  size_t buf = (size_t)N_IMG * HW * 4;          /* 32 MB each */
  float*    rowmin = (float*)(ws);
  float*    dc     = (float*)(ws + buf);
  unsigned* hist   = (unsigned*)(ws + 2 * buf);
  float*    thresh = (float*)(ws + 2 * buf + (size_t)N_IMG * NBINS * 4);
  unsigned* amax   = (unsigned*)(ws + 2 * buf + (size_t)N_IMG * NBINS * 4 + 64);

  k_init<<<(N_IMG * NBINS + 255) / 256, 256, 0, stream>>>(hist, amax);
  k_rowmin<<<N_IMG * H * (W / 256), 256, 0, stream>>>(I, rowmin);
  k_colmin_hist<<<N_IMG * (H / P2_TH) * (W / P2_TW), 256, 0, stream>>>(rowmin, dc, hist);
  k_thresh<<<N_IMG, 256, 0, stream>>>(hist, thresh);
  k_select<<<N_IMG * HW / 256, 256, 0, stream>>>(I, dc, thresh, amax);
  k_final<<<1, 32, 0, stream>>>(amax, out);
}